// QBWBlockGroupNorm_28381143892446
// MI455X (gfx1250) — compile-verified
//
#include <hip/hip_runtime.h>
#include <hip/hip_bf16.h>

typedef float v2f __attribute__((ext_vector_type(2)));
typedef float v8f __attribute__((ext_vector_type(8)));

#define GN_EPS 1e-5f

// ---------------------------------------------------------------------------
// Register-halving reduction: acc holds 8 rows x 16 lanes (one 16-lane group).
// After 8 shuffles (4+2+1+1), lane l holds sum/sumsq of row (l&7) over all 16
// lanes of its group. Valid for both lane groups (0-15, 16-31) independently.
// ---------------------------------------------------------------------------
__device__ __forceinline__ void reduce8rows(const v8f& acc, int lane,
                                            float& rs, float& rq) {
  float s0[8], q0[8];
#pragma unroll
  for (int j = 0; j < 8; ++j) {
    s0[j] = acc[j];
    q0[j] = acc[j] * acc[j];
  }
  const bool b0 = lane & 1, b1 = lane & 2, b2 = lane & 4;
  float s1[4], q1[4];
#pragma unroll
  for (int j = 0; j < 4; ++j) {
    float ms = b0 ? s0[2 * j + 1] : s0[2 * j];
    float os = b0 ? s0[2 * j] : s0[2 * j + 1];
    float mq = b0 ? q0[2 * j + 1] : q0[2 * j];
    float oq = b0 ? q0[2 * j] : q0[2 * j + 1];
    s1[j] = ms + __shfl_xor(os, 1, 32);
    q1[j] = mq + __shfl_xor(oq, 1, 32);
  }
  float s2[2], q2[2];
#pragma unroll
  for (int j = 0; j < 2; ++j) {
    float ms = b1 ? s1[2 * j + 1] : s1[2 * j];
    float os = b1 ? s1[2 * j] : s1[2 * j + 1];
    float mq = b1 ? q1[2 * j + 1] : q1[2 * j];
    float oq = b1 ? q1[2 * j] : q1[2 * j + 1];
    s2[j] = ms + __shfl_xor(os, 2, 32);
    q2[j] = mq + __shfl_xor(oq, 2, 32);
  }
  {
    float ms = b2 ? s2[1] : s2[0];
    float os = b2 ? s2[0] : s2[1];
    float mq = b2 ? q2[1] : q2[0];
    float oq = b2 ? q2[0] : q2[1];
    rs = ms + __shfl_xor(os, 4, 32);
    rq = mq + __shfl_xor(oq, 4, 32);
  }
  rs += __shfl_xor(rs, 8, 32);
  rq += __shfl_xor(rq, 8, 32);
}

// ---------------------------------------------------------------------------
// Fused 1x1-conv-as-GEMM + GroupNorm statistics.
//   Y[n, m, s] = sum_k W[m,k] * act(X[n,k,s])   (act = GN+ReLU when NORM)
// One wave computes a 16-wide S stripe for all MT m-tiles. Per K-chunk (KC),
// ALL A fragments (v2f -> global_load_b64) and all B fragments are staged
// into distinct registers first, then KI x MT WMMAs run back-to-back from
// registers (single wait, maximal memory-level parallelism).
// ---------------------------------------------------------------------------
template <int MT, int KC, bool NORM>
__global__ __launch_bounds__(32) void gemm_gn_kernel(
    const float* __restrict__ Wm,    // [MT*16, K] row-major
    const float* __restrict__ X,     // [NB, K, S] raw
    const float* __restrict__ nsc,   // [NB*K] per-(n,k) scale (NORM only)
    const float* __restrict__ nsh,   // [NB*K] per-(n,k) shift (NORM only)
    float* __restrict__ Y,           // [NB, MT*16, S] raw (pre-GN)
    float* __restrict__ ssum,        // [NB * MT*16]
    float* __restrict__ ssq,         // [NB * MT*16]
    int K, int S) {
  constexpr int KI = KC / 4;         // WMMA steps per chunk
  const int lane = threadIdx.x;      // 0..31
  const int col  = lane & 15;        // N within tile
  const int half = lane >> 4;        // 0/1
  const int s    = blockIdx.x * 16 + col;
  const int n    = blockIdx.y;
  const int M    = MT * 16;

  const float* Xn = X + (size_t)n * K * S + s;
  const float* Wl = Wm + col * K + 2 * half;  // + t*16*K + kb

  v8f acc[MT];
#pragma unroll
  for (int t = 0; t < MT; ++t) acc[t] = (v8f){0.f, 0.f, 0.f, 0.f, 0.f, 0.f, 0.f, 0.f};

  for (int kc = 0; kc < K; kc += KC) {
    // Stage A fragments for the whole chunk (b64 loads, distinct regs).
    v2f a[MT][KI];
#pragma unroll
    for (int t = 0; t < MT; ++t)
#pragma unroll
      for (int i = 0; i < KI; ++i)
        a[t][i] = *(const v2f*)(Wl + t * 16 * K + kc + 4 * i);

    // Stage B fragments for the whole chunk.
    v2f b[KI];
#pragma unroll
    for (int i = 0; i < KI; ++i) {
      const int kb = kc + 4 * i + 2 * half;
      b[i].x = Xn[(size_t)(kb + 0) * S];
      b[i].y = Xn[(size_t)(kb + 1) * S];
    }
    if (NORM) {
#pragma unroll
      for (int i = 0; i < KI; ++i) {
        const int kb = kc + 4 * i + 2 * half;
        b[i].x = fmaxf(fmaf(b[i].x, nsc[n * K + kb + 0], nsh[n * K + kb + 0]), 0.f);
        b[i].y = fmaxf(fmaf(b[i].y, nsc[n * K + kb + 1], nsh[n * K + kb + 1]), 0.f);
      }
    }

    // Dense WMMA burst, all operands in registers.
#pragma unroll
    for (int i = 0; i < KI; ++i)
#pragma unroll
      for (int t = 0; t < MT; ++t)
        acc[t] = __builtin_amdgcn_wmma_f32_16x16x4_f32(
            /*neg_a=*/false, a[t][i], /*neg_b=*/false, b[i],
            /*c_mod=*/(short)0, acc[t], /*reuse_a=*/false, /*reuse_b=*/false);
  }

  // Store raw output + accumulate per-row GN statistics.
#pragma unroll
  for (int t = 0; t < MT; ++t) {
#pragma unroll
    for (int j = 0; j < 8; ++j) {
      const int row = t * 16 + j + 8 * half;  // D layout: vgpr j -> M = j + 8*half
      Y[((size_t)n * M + row) * S + s] = acc[t][j];
    }
    float rs, rq;
    reduce8rows(acc[t], lane, rs, rq);
    if ((lane & 8) == 0) {
      const int row = t * 16 + (lane & 7) + 8 * half;
      atomicAdd(&ssum[n * M + row], rs);
      atomicAdd(&ssq[n * M + row], rq);
    }
  }
}

// ---------------------------------------------------------------------------
// Convert per-(n,c) sum/sum^2 into fused normalization scale/shift:
//   scale = gamma[c] * rsqrt(var+eps);  shift = beta[c] - mean*scale
// ---------------------------------------------------------------------------
__global__ __launch_bounds__(256) void finalize_gn_kernel(
    const float* __restrict__ ssum, const float* __restrict__ ssq,
    const float* __restrict__ gamma, const float* __restrict__ beta,
    float* __restrict__ scale, float* __restrict__ shift, int C, int NC,
    float invHW) {
  const int i = blockIdx.x * 256 + threadIdx.x;
  if (i >= NC) return;
  const float mean = ssum[i] * invHW;
  const float var  = ssq[i] * invHW - mean * mean;
  const float rstd = rsqrtf(var + GN_EPS);
  const int   c    = i % C;
  const float sc   = gamma[c] * rstd;
  scale[i] = sc;
  shift[i] = beta[c] - mean * sc;
}

// ---------------------------------------------------------------------------
// 3x3 depthwise conv; applies GN1+ReLU to inputs on the fly, writes raw
// output and accumulates GN2 statistics (full-wave reduce; HW=3136 is a
// multiple of 32, so each wave stays within one (n,c) plane).
// ---------------------------------------------------------------------------
__global__ __launch_bounds__(256) void dwconv_gn_kernel(
    const float* __restrict__ Yin, const float* __restrict__ sc,
    const float* __restrict__ sh, const float* __restrict__ Wd,  // [C*9]
    float* __restrict__ Yout, float* __restrict__ ssum, float* __restrict__ ssq,
    int C, int H, int Wdim) {
  const int idx = blockIdx.x * 256 + threadIdx.x;
  const int HW  = H * Wdim;
  const int np  = idx / HW;
  const int s   = idx % HW;
  const int p   = np % C;
  const int h   = s / Wdim, w = s % Wdim;

  const float scv = sc[np], shv = sh[np];
  const float* base = Yin + (size_t)np * HW;
  const float* wk   = Wd + p * 9;

  float acc = 0.f;
#pragma unroll
  for (int kh = -1; kh <= 1; ++kh) {
#pragma unroll
    for (int kw = -1; kw <= 1; ++kw) {
      const int hh = h + kh, ww = w + kw;
      if (hh >= 0 && hh < H && ww >= 0 && ww < Wdim) {
        float v = base[hh * Wdim + ww];
        v = fmaxf(fmaf(v, scv, shv), 0.f);
        acc = fmaf(v, wk[(kh + 1) * 3 + (kw + 1)], acc);
      }
    }
  }
  Yout[idx] = acc;

  float rs = acc, rq = acc * acc;
#pragma unroll
  for (int off = 1; off < 32; off <<= 1) {
    rs += __shfl_xor(rs, off, 32);
    rq += __shfl_xor(rq, off, 32);
  }
  if ((threadIdx.x & 31) == 0) {
    atomicAdd(&ssum[np], rs);
    atomicAdd(&ssq[np], rq);
  }
}

// out = GN3(y3) + GNs(ysc), fused scale/shift form.
__global__ __launch_bounds__(256) void final_add_kernel(
    const float* __restrict__ y3, const float* __restrict__ ysc,
    const float* __restrict__ sc3, const float* __restrict__ sh3,
    const float* __restrict__ scS, const float* __restrict__ shS,
    float* __restrict__ out, int HW) {
  const int idx = blockIdx.x * 256 + threadIdx.x;
  const int np  = idx / HW;
  out[idx] = fmaf(y3[idx], sc3[np], sh3[np]) + fmaf(ysc[idx], scS[np], shS[np]);
}

extern "C" void kernel_launch(void* const* d_in, const int* in_sizes, int n_in,
                              void* d_out, int out_size, void* d_ws, size_t ws_size,
                              hipStream_t stream) {
  (void)in_sizes; (void)n_in; (void)out_size; (void)ws_size;

  const float* x  = (const float*)d_in[0];
  const float* w1 = (const float*)d_in[1];
  const float* g1 = (const float*)d_in[2];
  const float* b1 = (const float*)d_in[3];
  const float* w2 = (const float*)d_in[4];
  const float* g2 = (const float*)d_in[5];
  const float* b2 = (const float*)d_in[6];
  const float* w3 = (const float*)d_in[7];
  const float* g3 = (const float*)d_in[8];
  const float* b3 = (const float*)d_in[9];
  const float* wS = (const float*)d_in[10];
  const float* gS = (const float*)d_in[11];
  const float* bS = (const float*)d_in[12];
  float* out = (float*)d_out;

  const int B = 64, Cin = 32, P = 192, Cout = 64, H = 56, W = 56;
  const int S = H * W;                       // 3136
  const size_t szY1 = (size_t)B * P * S;     // 38,535,168 floats
  const size_t szY3 = (size_t)B * Cout * S;  // 12,845,056 floats

  float* wsf = (float*)d_ws;
  float* y1  = wsf;                // [0, szY1)
  float* y2  = wsf + szY1;         // [szY1, 2*szY1)
  float* y3  = wsf;                // alias over y1 (y1 dead after dwconv)
  float* ysc = wsf + szY3;         // alias over y1 tail
  float* st  = wsf + 2 * szY1;     // stats block

  const int NC1 = B * P;           // 12288
  const int NC3 = B * Cout;        // 4096
  float* sum1 = st;               float* sq1 = sum1 + NC1;
  float* sum2 = sq1 + NC1;        float* sq2 = sum2 + NC1;
  float* sum3 = sq2 + NC1;        float* sq3 = sum3 + NC3;
  float* sumS = sq3 + NC3;        float* sqS = sumS + NC3;
  float* sc1  = sqS + NC3;        float* sh1 = sc1 + NC1;
  float* sc2  = sh1 + NC1;        float* sh2 = sc2 + NC1;
  float* sc3  = sh2 + NC1;        float* sh3 = sc3 + NC3;
  float* scS  = sh3 + NC3;        float* shS = scS + NC3;

  const size_t statBytes = (size_t)(4 * NC1 + 4 * NC3) * sizeof(float);
  hipMemsetAsync(st, 0, statBytes, stream);

  const float invHW = 1.0f / (float)S;
  const dim3 gGemm(S / 16, B);     // (196, 64) waves of 32

  // Stage 1: 1x1 expand (K=32, M=192 -> 12 m-tiles, KC=16) + GN1 stats
  gemm_gn_kernel<12, 16, false><<<gGemm, 32, 0, stream>>>(
      w1, x, nullptr, nullptr, y1, sum1, sq1, Cin, S);
  finalize_gn_kernel<<<(NC1 + 255) / 256, 256, 0, stream>>>(
      sum1, sq1, g1, b1, sc1, sh1, P, NC1, invHW);

  // Stage 2: depthwise 3x3 (consumes GN1+ReLU) + GN2 stats
  dwconv_gn_kernel<<<(int)(szY1 / 256), 256, 0, stream>>>(
      y1, sc1, sh1, w2, y2, sum2, sq2, P, H, W);
  finalize_gn_kernel<<<(NC1 + 255) / 256, 256, 0, stream>>>(
      sum2, sq2, g2, b2, sc2, sh2, P, NC1, invHW);

  // Stage 3: 1x1 project (K=192, M=64, KC=32), normalizes B operand on load
  gemm_gn_kernel<4, 32, true><<<gGemm, 32, 0, stream>>>(
      w3, y2, sc2, sh2, y3, sum3, sq3, P, S);
  // Shortcut: 1x1 (K=32, M=64, KC=32) from raw x
  gemm_gn_kernel<4, 32, false><<<gGemm, 32, 0, stream>>>(
      wS, x, nullptr, nullptr, ysc, sumS, sqS, Cin, S);

  finalize_gn_kernel<<<(NC3 + 255) / 256, 256, 0, stream>>>(
      sum3, sq3, g3, b3, sc3, sh3, Cout, NC3, invHW);
  finalize_gn_kernel<<<(NC3 + 255) / 256, 256, 0, stream>>>(
      sumS, sqS, gS, bS, scS, shS, Cout, NC3, invHW);

  // Final: GN3(y3) + GNs(ysc)
  final_add_kernel<<<(int)(szY3 / 256), 256, 0, stream>>>(
      y3, ysc, sc3, sh3, scS, shS, out, S);
}